// DTMS_89335319757368
// MI455X (gfx1250) — compile-verified
//
#include <hip/hip_runtime.h>
#include <math.h>

// ---------------------------------------------------------------------------
// DTMS forward for MI455X (gfx1250, wave32).
// fp32 end-to-end; dense layers use V_WMMA_F32_16X16X4_F32 (exact fp32 matrix
// pipe) with 16x64-per-wave register blocking and a 32x64 B panel staged in
// LDS. All buffers fit in the 192MB L2 -> compute/latency bound, not HBM.
// ---------------------------------------------------------------------------

typedef __attribute__((ext_vector_type(2))) float v2f;
typedef __attribute__((ext_vector_type(8))) float v8f;

#define DEVI __device__ __forceinline__

DEVI float silu_(float x) { return x / (1.0f + expf(-x)); }
DEVI float softplus_(float x) { return (x > 20.0f) ? x : log1pf(expf(x)); }

// -------------------------------- upsample ---------------------------------
// input1 (4,128,32,32) -> (4,128,64,64), align-corners bilinear (linspace 0..n-1)
__global__ void k_upsample(const float* __restrict__ in, float* __restrict__ out) {
  const int Cc = 128, Hi = 32, Wi = 32, Ho = 64, Wo = 64;
  int idx = blockIdx.x * blockDim.x + threadIdx.x;
  int wo = idx % Wo; int t = idx / Wo;
  int ho = t % Ho;   t /= Ho;
  int c  = t % Cc;   int b = t / Cc;
  float ph = ho * ((float)(Hi - 1) / (float)(Ho - 1));
  float pw = wo * ((float)(Wi - 1) / (float)(Wo - 1));
  int h0 = (int)floorf(ph); int h1 = min(h0 + 1, Hi - 1); float fh = ph - (float)h0;
  int w0 = (int)floorf(pw); int w1 = min(w0 + 1, Wi - 1); float fw = pw - (float)w0;
  const float* p = in + ((size_t)(b * Cc + c) * Hi) * Wi;
  float v00 = p[h0 * Wi + w0], v01 = p[h0 * Wi + w1];
  float v10 = p[h1 * Wi + w0], v11 = p[h1 * Wi + w1];
  float v0 = v00 * (1.f - fw) + v01 * fw;
  float v1 = v10 * (1.f - fw) + v11 * fw;
  out[idx] = v0 * (1.f - fh) + v1 * fh;
}

// ------------------------------ concat builders ----------------------------
// t1 NHWC (4,64,64,192): [x1up(NCHW 4,128,64,64) ; input2(NCHW 4,64,64,64)]
__global__ void k_build_t1(const float* __restrict__ x1up, const float* __restrict__ in2,
                           float* __restrict__ t1) {
  int idx = blockIdx.x * blockDim.x + threadIdx.x;
  int c = idx % 192; int t = idx / 192;
  int w = t % 64; t /= 64;
  int h = t % 64; int b = t / 64;
  float v = (c < 128) ? x1up[(((size_t)b * 128 + c) * 64 + h) * 64 + w]
                      : in2[(((size_t)b * 64 + (c - 128)) * 64 + h) * 64 + w];
  t1[idx] = v;
}

// t2 NHWC (4,64,128,64): width-concat of x1c (NCHW 4,64,64,64) and input2
__global__ void k_build_t2(const float* __restrict__ x1c, const float* __restrict__ in2,
                           float* __restrict__ t2) {
  int idx = blockIdx.x * blockDim.x + threadIdx.x;
  int c = idx % 64; int t = idx / 64;
  int w = t % 128; t /= 128;
  int h = t % 64;  int b = t / 64;
  float v = (w < 64) ? x1c[(((size_t)b * 64 + c) * 64 + h) * 64 + w]
                     : in2[(((size_t)b * 64 + c) * 64 + h) * 64 + (w - 64)];
  t2[idx] = v;
}

// -------------------------------- layer norm -------------------------------
// one block per row, C <= 1024
__global__ void k_layernorm(const float* __restrict__ x, const float* __restrict__ g,
                            const float* __restrict__ bta, float* __restrict__ y, int C) {
  __shared__ float sdata[256];
  int row = blockIdx.x;
  const float* xr = x + (size_t)row * C;
  float* yr = y + (size_t)row * C;
  float s = 0.f, s2 = 0.f;
  for (int c = threadIdx.x; c < C; c += blockDim.x) { float v = xr[c]; s += v; s2 += v * v; }
  sdata[threadIdx.x] = s; __syncthreads();
  for (int o = blockDim.x >> 1; o > 0; o >>= 1) {
    if ((int)threadIdx.x < o) sdata[threadIdx.x] += sdata[threadIdx.x + o];
    __syncthreads();
  }
  float mean = sdata[0] / (float)C; __syncthreads();
  sdata[threadIdx.x] = s2; __syncthreads();
  for (int o = blockDim.x >> 1; o > 0; o >>= 1) {
    if ((int)threadIdx.x < o) sdata[threadIdx.x] += sdata[threadIdx.x + o];
    __syncthreads();
  }
  float var = sdata[0] / (float)C - mean * mean;
  float inv = rsqrtf(var + 1e-5f);
  for (int c = threadIdx.x; c < C; c += blockDim.x)
    yr[c] = (xr[c] - mean) * inv * g[c] + bta[c];
}

// ------------------------------- WMMA fp32 GEMM ----------------------------
// D = A(MxK) * B(KxN) [+bias] [+act] [+resid], row-major, fp32.
// Block = 4 waves. Block tile 64(M) x 64(N); each wave owns 16(M) x 64(N)
// (4 v8f accumulators, A-frag reused x4). B panel (32K x 64N) staged in LDS
// and shared by all 4 waves. Requires M%64==0, N%64==0, K%32==0.
// EPI: 0 none, 1 bias+GELU(exact), 2 bias+resid, 3 resid
#define GEMM_KT 32
#define GEMM_BS 72   // padded LDS row stride: half0 (banks idx) / half1 (idx+16) disjoint
template <int EPI>
__global__ void k_gemm_wmma(const float* __restrict__ A, const float* __restrict__ Bm,
                            const float* __restrict__ bias, const float* __restrict__ resid,
                            float* __restrict__ D, int M, int N, int K) {
  __shared__ float Bs[GEMM_KT * GEMM_BS];
  int wave = threadIdx.x >> 5;
  int lane = threadIdx.x & 31;
  int half = lane >> 4;   // 0: K pair {0,1}, M 0..7 ; 1: K pair {2,3}, M 8..15
  int idx  = lane & 15;
  int tm  = (blockIdx.x * 4 + wave) * 16;
  int tn0 = blockIdx.y * 64;

  v8f acc[4];
#pragma unroll
  for (int g2 = 0; g2 < 4; ++g2) acc[g2] = (v8f){};

  for (int kp = 0; kp < K; kp += GEMM_KT) {
    __syncthreads();
    // cooperative B panel load: rows kp..kp+31, cols tn0..tn0+63 (coalesced)
    for (int i = threadIdx.x; i < GEMM_KT * 64; i += 128) {
      int kk = i >> 6; int nn = i & 63;
      Bs[kk * GEMM_BS + nn] = Bm[(size_t)(kp + kk) * N + tn0 + nn];
    }
    __syncthreads();
    const float* Ap = A + (size_t)(tm + idx) * K + kp + 2 * half;
#pragma unroll
    for (int ks = 0; ks < GEMM_KT; ks += 4) {
      v2f av;
      av.x = Ap[ks];
      av.y = Ap[ks + 1];
      const float* bp0 = &Bs[(ks + 2 * half) * GEMM_BS + idx];
#pragma unroll
      for (int g2 = 0; g2 < 4; ++g2) {
        v2f bv;
        bv.x = bp0[g2 * 16];
        bv.y = bp0[g2 * 16 + GEMM_BS];
        acc[g2] = __builtin_amdgcn_wmma_f32_16x16x4_f32(false, av, false, bv,
                                                        (short)0, acc[g2], false, false);
      }
    }
  }

#pragma unroll
  for (int g2 = 0; g2 < 4; ++g2) {
    int n = tn0 + g2 * 16 + idx;
    float bi = (EPI == 1 || EPI == 2) ? bias[n] : 0.0f;
#pragma unroll
    for (int v = 0; v < 8; ++v) {
      int m = tm + v + 8 * half;
      float x = acc[g2][v];
      if (EPI == 1) { x += bi; x = 0.5f * x * (1.0f + erff(x * 0.70710678118654752f)); }
      else if (EPI == 2) { x += bi + resid[(size_t)m * N + n]; }
      else if (EPI == 3) { x += resid[(size_t)m * N + n]; }
      D[(size_t)m * N + n] = x;
    }
  }
}

// ----------------------- depthwise conv 3x3 + SiLU -------------------------
// in: NHWC (B,H,W,C) ; out: NCHW (B,C,H,W) ; SAME padding
__global__ void k_dwconv_silu(const float* __restrict__ xin, const float* __restrict__ wt,
                              float* __restrict__ xc, int Bn, int Hh, int Ww, int Cc) {
  int idx = blockIdx.x * blockDim.x + threadIdx.x;
  int w = idx % Ww; int t = idx / Ww;
  int h = t % Hh; t /= Hh;
  int c = t % Cc; int b = t / Cc;
  float acc = 0.f;
#pragma unroll
  for (int kh = 0; kh < 3; ++kh) {
    int ih = h + kh - 1;
    if (ih < 0 || ih >= Hh) continue;
#pragma unroll
    for (int kw = 0; kw < 3; ++kw) {
      int iw = w + kw - 1;
      if (iw < 0 || iw >= Ww) continue;
      acc += xin[(((size_t)b * Hh + ih) * Ww + iw) * Cc + c] * wt[c * 9 + kh * 3 + kw];
    }
  }
  xc[idx] = silu_(acc);
}

// direction -> physical hw index for scan position l
DEVI int dir_phys(int k, int l, int Hh, int Ww) {
  int L = Hh * Ww;
  int ll = (k >= 2) ? (L - 1 - l) : l;
  if (k & 1) { int hh = ll % Hh; int ww = ll / Hh; return hh * Ww + ww; }
  return ll;
}

// ----------------------------- x_proj (skinny) -----------------------------
// x_dbl[b,k,c,l] = sum_d x_proj_w[k,c,d] * xs[b,k,d,l]; xs gathered from xconv NCHW
template <int CN>
__global__ void k_xproj(const float* __restrict__ xc, const float* __restrict__ xw,
                        float* __restrict__ xdbl, int Bn, int Din, int Hh, int Ww) {
  int L = Hh * Ww;
  int idx = blockIdx.x * blockDim.x + threadIdx.x;
  int l = idx % L; int t = idx / L;
  int k = t % 4;  int b = t / 4;
  int pl = dir_phys(k, l, Hh, Ww);
  float acc[CN];
#pragma unroll
  for (int c = 0; c < CN; ++c) acc[c] = 0.f;
  const float* xp = xc + (size_t)b * Din * L + pl;
  for (int d = 0; d < Din; ++d) {
    float xv = xp[(size_t)d * L];
#pragma unroll
    for (int c = 0; c < CN; ++c)
      acc[c] += xw[((size_t)k * CN + c) * Din + d] * xv;
  }
#pragma unroll
  for (int c = 0; c < CN; ++c)
    xdbl[(((size_t)b * 4 + k) * CN + c) * L + l] = acc[c];
}

// ------------------------------ selective scan -----------------------------
// One thread per (b, d): runs all 4 directions serially and accumulates the
// four directional outputs at the same physical index (deterministic, no atomics).
// y stored NHWC-style (B, L, Din).
template <int RR>
__global__ void k_scan(const float* __restrict__ xc, const float* __restrict__ xdbl,
                       const float* __restrict__ dtw, const float* __restrict__ dtb,
                       const float* __restrict__ Alog, const float* __restrict__ Dp,
                       float* __restrict__ y, int Bn, int Din, int Hh, int Ww) {
  const int Cn = RR + 2;
  int L = Hh * Ww;
  int tid = blockIdx.x * blockDim.x + threadIdx.x;
  if (tid >= Bn * Din) return;
  int d = tid % Din; int b = tid / Din;
  float* yp = y + (size_t)b * L * Din + d;
  for (int k = 0; k < 4; ++k) {
    float w[RR];
#pragma unroll
    for (int r = 0; r < RR; ++r) w[r] = dtw[((size_t)k * Din + d) * RR + r];
    float bias = dtb[k * Din + d];
    float Av = -expf(Alog[k * Din + d]);
    float Dv = Dp[k * Din + d];
    const float* xdp = xdbl + (size_t)(b * 4 + k) * Cn * L;
    const float* xcp = xc + ((size_t)b * Din + d) * L;
    float hstate = 0.f;
    for (int j = 0; j < L; ++j) {
      int pl = dir_phys(k, j, Hh, Ww);
      float xv = xcp[pl];
      float dt = bias;
#pragma unroll
      for (int r = 0; r < RR; ++r) dt += w[r] * xdp[(size_t)r * L + j];
      dt = softplus_(dt);
      float Bsv = xdp[(size_t)RR * L + j];
      float Csv = xdp[(size_t)(RR + 1) * L + j];
      float a = expf(dt * Av);
      hstate = a * hstate + dt * Bsv * xv;
      float yv = Csv * hstate + Dv * xv;
      if (k == 0) yp[(size_t)pl * Din] = yv;
      else        yp[(size_t)pl * Din] += yv;
    }
  }
}

// --------------------------- conv 3x3 + BN + ReLU --------------------------
// NCHW in, NCHW out, SAME
__global__ void k_conv_cba_nchw(const float* __restrict__ x, const float* __restrict__ wt,
                                const float* __restrict__ g, const float* __restrict__ bb,
                                const float* __restrict__ mn, const float* __restrict__ vr,
                                float* __restrict__ y, int Bn, int Ci, int Co, int Hh, int Ww) {
  int idx = blockIdx.x * blockDim.x + threadIdx.x;
  int w = idx % Ww; int t = idx / Ww;
  int h = t % Hh; t /= Hh;
  int co = t % Co; int b = t / Co;
  float acc = 0.f;
  for (int ci = 0; ci < Ci; ++ci) {
    const float* xp = x + (((size_t)b * Ci + ci) * Hh) * Ww;
    const float* wp = wt + ((size_t)co * Ci + ci) * 9;
#pragma unroll
    for (int kh = 0; kh < 3; ++kh) {
      int ih = h + kh - 1;
      if (ih < 0 || ih >= Hh) continue;
#pragma unroll
      for (int kw = 0; kw < 3; ++kw) {
        int iw = w + kw - 1;
        if (iw < 0 || iw >= Ww) continue;
        acc += xp[ih * Ww + iw] * wp[kh * 3 + kw];
      }
    }
  }
  float scale = g[co] * rsqrtf(vr[co] + 1e-5f);
  y[idx] = fmaxf((acc - mn[co]) * scale + bb[co], 0.f);
}

// fused concat(t1out NHWC 192ch, t2 left 64ch, t2 right 64ch) -> conv cbr2 -> NCHW
__global__ void k_conv_cba_cat(const float* __restrict__ t1o, const float* __restrict__ t2o,
                               const float* __restrict__ wt, const float* __restrict__ g,
                               const float* __restrict__ bb, const float* __restrict__ mn,
                               const float* __restrict__ vr, float* __restrict__ y) {
  const int Ci = 320, Co = 64, Hh = 64, Ww = 64;
  int idx = blockIdx.x * blockDim.x + threadIdx.x;
  int w = idx % Ww; int t = idx / Ww;
  int h = t % Hh; t /= Hh;
  int co = t % Co; int b = t / Co;
  float acc = 0.f;
  for (int ci = 0; ci < Ci; ++ci) {
    const float* wp = wt + ((size_t)co * Ci + ci) * 9;
#pragma unroll
    for (int kh = 0; kh < 3; ++kh) {
      int ih = h + kh - 1;
      if (ih < 0 || ih >= Hh) continue;
#pragma unroll
      for (int kw = 0; kw < 3; ++kw) {
        int iw = w + kw - 1;
        if (iw < 0 || iw >= Ww) continue;
        float xv;
        if (ci < 192)
          xv = t1o[(((size_t)b * 64 + ih) * 64 + iw) * 192 + ci];
        else if (ci < 256)
          xv = t2o[(((size_t)b * 64 + ih) * 128 + iw) * 64 + (ci - 192)];
        else
          xv = t2o[(((size_t)b * 64 + ih) * 128 + (iw + 64)) * 64 + (ci - 256)];
        acc += xv * wp[kh * 3 + kw];
      }
    }
  }
  float scale = g[co] * rsqrtf(vr[co] + 1e-5f);
  y[idx] = fmaxf((acc - mn[co]) * scale + bb[co], 0.f);
}

// ------------------------------- host plumbing -----------------------------
struct SsmParams {
  const float *norm_g, *norm_b, *in_proj, *conv_w, *x_proj_w, *dt_proj_w, *dt_bias,
              *A_log, *D, *out_norm_g, *out_norm_b, *out_proj, *norm2_g, *norm2_b,
              *fc1_w, *fc1_b, *fc2_w, *fc2_b;
};
struct CbaParams { const float *w, *g, *b, *mean, *var; };

static SsmParams load_ssm(void* const* d_in, int base) {
  SsmParams p;
  p.norm_g = (const float*)d_in[base + 0];  p.norm_b = (const float*)d_in[base + 1];
  p.in_proj = (const float*)d_in[base + 2]; p.conv_w = (const float*)d_in[base + 3];
  p.x_proj_w = (const float*)d_in[base + 4]; p.dt_proj_w = (const float*)d_in[base + 5];
  p.dt_bias = (const float*)d_in[base + 6]; p.A_log = (const float*)d_in[base + 7];
  p.D = (const float*)d_in[base + 8];       p.out_norm_g = (const float*)d_in[base + 9];
  p.out_norm_b = (const float*)d_in[base + 10]; p.out_proj = (const float*)d_in[base + 11];
  p.norm2_g = (const float*)d_in[base + 12]; p.norm2_b = (const float*)d_in[base + 13];
  p.fc1_w = (const float*)d_in[base + 14];  p.fc1_b = (const float*)d_in[base + 15];
  p.fc2_w = (const float*)d_in[base + 16];  p.fc2_b = (const float*)d_in[base + 17];
  return p;
}
static CbaParams load_cba(void* const* d_in, int base) {
  CbaParams p;
  p.w = (const float*)d_in[base + 0]; p.g = (const float*)d_in[base + 1];
  p.b = (const float*)d_in[base + 2]; p.mean = (const float*)d_in[base + 3];
  p.var = (const float*)d_in[base + 4];
  return p;
}

// full VSS block: xout = vss_block(xin); xin/xout NHWC (Bn,Hh,Ww,d); may alias.
static void run_vss(const SsmParams& p, const float* xin, float* xout,
                    int Bn, int Hh, int Ww, int d, int Din, int R,
                    float* SA, float* SB, float* SC, float* SD, float* SE, float* SF,
                    hipStream_t s) {
  const int L = Hh * Ww;
  const int M = Bn * L;
  const int tpb = 256;

  // 1. t_norm = LN(xin)  -> SD
  k_layernorm<<<M, tpb, 0, s>>>(xin, p.norm_g, p.norm_b, SD, d);
  // 2. xinproj = t_norm @ in_proj  -> SA  (M x Din)
  {
    dim3 g(M / 64, Din / 64);
    k_gemm_wmma<0><<<g, 128, 0, s>>>(SD, p.in_proj, nullptr, nullptr, SA, M, Din, d);
  }
  // 3. depthwise conv + silu : NHWC -> NCHW, -> SB
  k_dwconv_silu<<<(Bn * Din * L) / tpb, tpb, 0, s>>>(SA, p.conv_w, SB, Bn, Hh, Ww, Din);
  // 4. x_dbl -> SF
  if (R == 12) k_xproj<14><<<(Bn * 4 * L) / tpb, tpb, 0, s>>>(SB, p.x_proj_w, SF, Bn, Din, Hh, Ww);
  else         k_xproj<6> <<<(Bn * 4 * L) / tpb, tpb, 0, s>>>(SB, p.x_proj_w, SF, Bn, Din, Hh, Ww);
  // 5. selective scan -> y in SC (B, L, Din)
  {
    int tot = Bn * Din;
    int blocks = (tot + 127) / 128;
    if (R == 12) k_scan<12><<<blocks, 128, 0, s>>>(SB, SF, p.dt_proj_w, p.dt_bias, p.A_log, p.D, SC, Bn, Din, Hh, Ww);
    else         k_scan<4> <<<blocks, 128, 0, s>>>(SB, SF, p.dt_proj_w, p.dt_bias, p.A_log, p.D, SC, Bn, Din, Hh, Ww);
  }
  // 6. out_norm -> SD
  k_layernorm<<<M, tpb, 0, s>>>(SC, p.out_norm_g, p.out_norm_b, SD, Din);
  // 7. xres = xin + ynorm @ out_proj -> SE (M x d)
  {
    dim3 g(M / 64, d / 64);
    k_gemm_wmma<3><<<g, 128, 0, s>>>(SD, p.out_proj, nullptr, xin, SE, M, d, Din);
  }
  // 8. h = LN2(xres) -> SD
  k_layernorm<<<M, tpb, 0, s>>>(SE, p.norm2_g, p.norm2_b, SD, d);
  // 9. fcmid = gelu(h @ fc1 + b1) -> SA (M x 4d)
  {
    dim3 g(M / 64, (4 * d) / 64);
    k_gemm_wmma<1><<<g, 128, 0, s>>>(SD, p.fc1_w, p.fc1_b, nullptr, SA, M, 4 * d, d);
  }
  // 10. xout = xres + fcmid @ fc2 + b2
  {
    dim3 g(M / 64, d / 64);
    k_gemm_wmma<2><<<g, 128, 0, s>>>(SA, p.fc2_w, p.fc2_b, SE, xout, M, d, 4 * d);
  }
}

extern "C" void kernel_launch(void* const* d_in, const int* in_sizes, int n_in,
                              void* d_out, int out_size, void* d_ws, size_t ws_size,
                              hipStream_t stream) {
  (void)in_sizes; (void)n_in; (void)out_size; (void)ws_size;
  const float* input1 = (const float*)d_in[0];
  const float* input2 = (const float*)d_in[1];
  SsmParams ssm1 = load_ssm(d_in, 2);
  SsmParams ssm2 = load_ssm(d_in, 20);
  CbaParams cbr  = load_cba(d_in, 38);
  CbaParams cbr2 = load_cba(d_in, 43);
  CbaParams cbr3 = load_cba(d_in, 48);

  // workspace layout (float elems), total ~41.9M floats = ~168MB
  float* ws = (float*)d_ws;
  float* T1 = ws;                    // 3,145,728  t1 NHWC / t1out
  float* T2 = T1 + 3145728;          // 2,097,152  t2in NHWC / t2out
  float* SA = T2 + 2097152;          // 12,582,912 in_proj out / fc mid
  float* SB = SA + 12582912;         // 6,291,456  x1up (early) / xconv
  float* SC = SB + 6291456;          // 6,291,456  scan output y
  float* SD = SC + 6291456;          // 6,291,456  layernorm scratch
  float* SE = SD + 6291456;          // 3,145,728  residual xres
  float* SF = SE + 3145728;          // 1,048,576  x1c (early) / x_dbl
  float* SG = SF + 1048576;          // 1,048,576  cbr2 output

  const int tpb = 256;
  // 1) x1up = bilinear(input1) -> SB (NCHW 4,128,64,64)
  k_upsample<<<2097152 / tpb, tpb, 0, stream>>>(input1, SB);
  // 2) t1 = concat(x1up, input2) NHWC -> T1
  k_build_t1<<<3145728 / tpb, tpb, 0, stream>>>(SB, input2, T1);
  // 3) x1c = cba(x1up, cbr) -> SF (NCHW 4,64,64,64)
  k_conv_cba_nchw<<<1048576 / tpb, tpb, 0, stream>>>(SB, cbr.w, cbr.g, cbr.b, cbr.mean, cbr.var,
                                                     SF, 4, 128, 64, 64, 64);
  // 4) t2in = width-concat(x1c, input2) NHWC -> T2
  k_build_t2<<<2097152 / tpb, tpb, 0, stream>>>(SF, input2, T2);
  // 5) vss block 1: d=192, Din=384, R=12, H=64, W=64
  run_vss(ssm1, T1, T1, 4, 64, 64, 192, 384, 12, SA, SB, SC, SD, SE, SF, stream);
  // 6) vss block 2: d=64, Din=128, R=4, H=64, W=128
  run_vss(ssm2, T2, T2, 4, 64, 128, 64, 128, 4, SA, SB, SC, SD, SE, SF, stream);
  // 7) t = cba(concat(t1out, t2 halves), cbr2) -> SG (NCHW 4,64,64,64)
  k_conv_cba_cat<<<1048576 / tpb, tpb, 0, stream>>>(T1, T2, cbr2.w, cbr2.g, cbr2.b,
                                                    cbr2.mean, cbr2.var, SG);
  // 8) out = cba(t, cbr3) -> d_out (NCHW 4,64,64,64)
  k_conv_cba_nchw<<<1048576 / tpb, tpb, 0, stream>>>(SG, cbr3.w, cbr3.g, cbr3.b, cbr3.mean,
                                                     cbr3.var, (float*)d_out, 4, 64, 64, 64, 64);
}